// SpiralTwistedInteractionHead_56788057588317
// MI455X (gfx1250) — compile-verified
//
#include <hip/hip_runtime.h>
#include <math.h>

typedef __attribute__((ext_vector_type(2))) float v2f;
typedef __attribute__((ext_vector_type(8))) float v8f;

#define C_RED   32
#define IN_CH   256
#define IMG     64
#define HW      4096          // 64*64
#define NB      16            // batch
#define OUT_CH  560           // 32 + 528
#define BN_EPS  1e-5f
#define NORM_EPS 1e-6f

// ---------------------------------------------------------------------------
// Kernel 1: z = relu(bn(1x1conv(x))) via V_WMMA_F32_16X16X4_F32
// GEMM view: C[M=pixels][N=32] = sum_k A[p][k]*B[k][n],
//   A[p][k] = x[b][k][p], B[k][n] = w_reduce[n][k]
// One wave handles 64 pixels (4 M-tiles of 16) x full N=32 (2 N-tiles).
// ---------------------------------------------------------------------------
__global__ __launch_bounds__(256) void k1_reduce_bn_relu(
    const float* __restrict__ x, const float* __restrict__ w,
    const float* __restrict__ gamma, const float* __restrict__ beta,
    const float* __restrict__ mean, const float* __restrict__ var,
    float* __restrict__ z)
{
  const int lane   = threadIdx.x & 31;
  const int gwave  = blockIdx.x * 8 + (threadIdx.x >> 5);
  const int P0     = gwave * 64;            // first pixel of this wave
  const int m      = lane & 15;             // row within M-tile
  const int n      = lane & 15;             // col within N-tile
  const int khalf  = (lane >> 4) * 2;       // lanes 16-31 carry K+2, K+3
  const int b      = P0 >> 12;              // / 4096 (tiles never cross images)
  const int pimg   = P0 & 4095;

  const float* xb = x + (size_t)b * (IN_CH * HW) + pimg;
  const float* w0 = w + (size_t)(n)      * IN_CH;   // B col n,    N-tile 0
  const float* w1 = w + (size_t)(n + 16) * IN_CH;   // B col n+16, N-tile 1

  v8f acc[2][4];
  #pragma unroll
  for (int nt = 0; nt < 2; ++nt)
    #pragma unroll
    for (int t = 0; t < 4; ++t)
      acc[nt][t] = (v8f){};

  for (int k0 = 0; k0 < IN_CH; k0 += 4) {
    const int kk = k0 + khalf;
    v2f bf0, bf1;
    bf0.x = w0[kk];  bf0.y = w0[kk + 1];
    bf1.x = w1[kk];  bf1.y = w1[kk + 1];
    const float* ap = xb + (size_t)kk * HW + m;
    #pragma unroll
    for (int t = 0; t < 4; ++t) {
      v2f af;
      af.x = ap[t * 16];
      af.y = ap[t * 16 + HW];
      acc[0][t] = __builtin_amdgcn_wmma_f32_16x16x4_f32(
          false, af, false, bf0, (short)0, acc[0][t], false, false);
      acc[1][t] = __builtin_amdgcn_wmma_f32_16x16x4_f32(
          false, af, false, bf1, (short)0, acc[1][t], false, false);
    }
  }

  // C/D layout: vgpr r, lanes 0-15 -> M=r ; lanes 16-31 -> M=8+r ; N = lane&15
  const int mrow = (lane >> 4) * 8;
  float* zb = z + (size_t)b * (C_RED * HW);
  #pragma unroll
  for (int nt = 0; nt < 2; ++nt) {
    const int c = nt * 16 + n;
    const float inv = gamma[c] / sqrtf(var[c] + BN_EPS);
    const float bb  = beta[c] - mean[c] * inv;
    #pragma unroll
    for (int t = 0; t < 4; ++t) {
      #pragma unroll
      for (int r = 0; r < 8; ++r) {
        const int p = pimg + t * 16 + mrow + r;
        const float v = acc[nt][t][r] * inv + bb;
        zb[(size_t)c * HW + p] = fmaxf(v, 0.0f);
      }
    }
  }
}

// ---------------------------------------------------------------------------
// Kernel 2: depthwise 3x3 (SAME, zero pad) + scale, dual L2-normalize,
// emit z_norm (32 ch) and upper-triangular pair products (528 ch).
// Tile: 32x8 pixels, halo in LDS (34*10*32 f32 = 43.5 KB).
// ---------------------------------------------------------------------------
#define TX 32
#define TY 8

__global__ __launch_bounds__(256) void k2_dw_norm_pairs(
    const float* __restrict__ z, const float* __restrict__ dww,
    const float* __restrict__ scale, float* __restrict__ out)
{
  __shared__ float tile[C_RED][TY + 2][TX + 2];

  const int x0 = blockIdx.x * TX;
  const int y0 = blockIdx.y * TY;
  const int b  = blockIdx.z;
  const int tid = threadIdx.x;
  const float* zb = z + (size_t)b * (C_RED * HW);

  // cooperative halo load, zero outside the image
  for (int i = tid; i < C_RED * (TY + 2) * (TX + 2); i += 256) {
    const int c  = i / ((TY + 2) * (TX + 2));
    const int r  = i % ((TY + 2) * (TX + 2));
    const int yy = r / (TX + 2);
    const int xx = r % (TX + 2);
    const int gy = y0 + yy - 1;
    const int gx = x0 + xx - 1;
    float v = 0.0f;
    if (gy >= 0 && gy < IMG && gx >= 0 && gx < IMG)
      v = zb[(size_t)c * HW + gy * IMG + gx];
    tile[c][yy][xx] = v;
  }
  __syncthreads();

  const int tx = tid & (TX - 1);
  const int ty = tid / TX;

  float zn[C_RED], tw[C_RED];
  float nz = 0.0f, nt = 0.0f;
  #pragma unroll
  for (int c = 0; c < C_RED; ++c) {
    const float* wv = dww + c * 9;   // uniform -> scalarized loads
    float s =
        wv[0] * tile[c][ty    ][tx] + wv[1] * tile[c][ty    ][tx + 1] + wv[2] * tile[c][ty    ][tx + 2]
      + wv[3] * tile[c][ty + 1][tx] + wv[4] * tile[c][ty + 1][tx + 1] + wv[5] * tile[c][ty + 1][tx + 2]
      + wv[6] * tile[c][ty + 2][tx] + wv[7] * tile[c][ty + 2][tx + 1] + wv[8] * tile[c][ty + 2][tx + 2];
    s *= scale[c];
    const float zc = tile[c][ty + 1][tx + 1];
    zn[c] = zc;
    tw[c] = s;
    nz += zc * zc;
    nt += s * s;
  }
  const float inz = 1.0f / fmaxf(sqrtf(nz), NORM_EPS);
  const float itw = 1.0f / fmaxf(sqrtf(nt), NORM_EPS);
  #pragma unroll
  for (int c = 0; c < C_RED; ++c) { zn[c] *= inz; tw[c] *= itw; }

  float* ob = out + (size_t)b * ((size_t)OUT_CH * HW)
                  + (size_t)(y0 + ty) * IMG + (x0 + tx);
  // z_norm channels 0..31
  #pragma unroll
  for (int c = 0; c < C_RED; ++c)
    ob[(size_t)c * HW] = zn[c];
  // triu pairs (row-major i<=j), channels 32..559 — fully unrolled so
  // zn[]/tw[] stay in VGPRs (compile-time indices, no scratch)
  #pragma unroll
  for (int i = 0; i < C_RED; ++i) {
    const int base = C_RED + i * C_RED - (i * (i - 1)) / 2 - i; // ch of pair (i,j) = base + j
    #pragma unroll
    for (int j = 0; j < C_RED; ++j) {
      if (j >= i)
        ob[(size_t)(base + j) * HW] = zn[i] * tw[j];
    }
  }
}

extern "C" void kernel_launch(void* const* d_in, const int* in_sizes, int n_in,
                              void* d_out, int out_size, void* d_ws, size_t ws_size,
                              hipStream_t stream) {
  (void)in_sizes; (void)n_in; (void)out_size; (void)ws_size;
  const float* x     = (const float*)d_in[0];
  const float* w     = (const float*)d_in[1];
  const float* gamma = (const float*)d_in[2];
  const float* beta  = (const float*)d_in[3];
  const float* mean  = (const float*)d_in[4];
  const float* var   = (const float*)d_in[5];
  const float* dww   = (const float*)d_in[6];
  const float* scale = (const float*)d_in[7];
  float* out = (float*)d_out;
  float* z   = (float*)d_ws;  // 16*32*64*64 f32 = 8.39 MB scratch

  // 65536 pixels / 64 per wave = 1024 waves = 128 blocks * 8 waves
  k1_reduce_bn_relu<<<128, 256, 0, stream>>>(x, w, gamma, beta, mean, var, z);
  dim3 g2(IMG / TX, IMG / TY, NB);
  k2_dw_norm_pairs<<<g2, 256, 0, stream>>>(z, dww, scale, out);
}